// BasicCGInducer_85787676770606
// MI455X (gfx1250) — compile-verified
//
#include <hip/hip_runtime.h>
#include <hip/hip_bf16.h>

typedef __attribute__((ext_vector_type(2))) float v2f;
typedef __attribute__((ext_vector_type(8))) float v8f;

#define NF    36
#define CDIM  2596
#define DD    64
#define VV    32000
#define NEGV  (-1.0e9f)

#define MTILES     163      // ceil(2596/16)
#define NCHUNK     10
#define CT_PER_CHUNK 200    // 32000/16/NCHUNK

// ---- workspace layout (float offsets) ----
#define LSE_OFF    0                    // 2608
#define SPLIT1_OFF 2608                 // 2608
#define GA_OFF     5216                 // 1296
#define GB_OFF     6512                 // 1296
#define ROOT_OFF   7808                 // 64
#define RULE_OFF   7872                 // 36*72 = 2592
#define PM_OFF     10464                // 2608*NCHUNK = 26080
#define PS_OFF     36544                // 26080
#define BETA_OFF   62624                // 1024*2596 = 2658304
#define CHART_OFF  2720928              // 33*32*32*36 = 1216512  (total 3937440 f = 15.75MB)

#define CIDX(L,s,b,r) ((((size_t)(L)*32 + (size_t)(s))*32 + (size_t)(b))*36 + (size_t)(r))

// ---------------------------------------------------------------------------
// K0: root log-softmax (masked, only [:36] needed) + rule log-softmax (36x72)
// ---------------------------------------------------------------------------
__global__ void k_small(const float* __restrict__ rootW, const float* __restrict__ rootb,
                        const float* __restrict__ ruleW, const float* __restrict__ ruleb,
                        float* __restrict__ ws) {
  const int t = threadIdx.x;               // 128 threads
  __shared__ float red[128];
  float lm = -3.0e38f;
  for (int i = t; i < CDIM; i += 128) {
    float v = ((i < 4) ? 0.f : NEGV) + rootW[i] + rootb[i];
    lm = fmaxf(lm, v);
  }
  red[t] = lm; __syncthreads();
  for (int st = 64; st > 0; st >>= 1) { if (t < st) red[t] = fmaxf(red[t], red[t+st]); __syncthreads(); }
  const float gm = red[0]; __syncthreads();
  float lsum = 0.f;
  for (int i = t; i < CDIM; i += 128) {
    float v = ((i < 4) ? 0.f : NEGV) + rootW[i] + rootb[i];
    lsum += __expf(v - gm);
  }
  red[t] = lsum; __syncthreads();
  for (int st = 64; st > 0; st >>= 1) { if (t < st) red[t] += red[t+st]; __syncthreads(); }
  const float lse = gm + __logf(red[0]);
  if (t < NF)
    ws[ROOT_OFF + t] = ((t < 4) ? 0.f : NEGV) + rootW[t] + rootb[t] - lse;
  // rule_scores: row-wise log-softmax over 72
  if (t < NF) {
    float rm = -3.0e38f;
    for (int j = 0; j < 72; ++j) rm = fmaxf(rm, ruleW[t*72 + j] + ruleb[j]);
    float rs = 0.f;
    for (int j = 0; j < 72; ++j) rs += __expf(ruleW[t*72 + j] + ruleb[j] - rm);
    const float rl = rm + __logf(rs);
    for (int j = 0; j < 72; ++j) ws[RULE_OFF + t*72 + j] = ruleW[t*72 + j] + ruleb[j] - rl;
  }
}

// ---------------------------------------------------------------------------
// K1: MLP over nt_emb rows -> split log-softmax; fold G_larg/G_rarg tables.
// One thread per row; activations in LDS (65-stride: conflict-free).
// ---------------------------------------------------------------------------
__global__ void k_mlp(const float* __restrict__ nt,
                      const float* __restrict__ sW1, const float* __restrict__ sb1,
                      const float* __restrict__ r1W1, const float* __restrict__ r1b1,
                      const float* __restrict__ r1W2, const float* __restrict__ r1b2,
                      const float* __restrict__ r2W1, const float* __restrict__ r2b1,
                      const float* __restrict__ r2W2, const float* __restrict__ r2b2,
                      const float* __restrict__ sW2, const float* __restrict__ sb2,
                      float* __restrict__ ws) {
  __shared__ float se[64][65], sh[64][65], st_[64][65];
  const int t = threadIdx.x;
  const int c = blockIdx.x * 64 + t;
  if (c >= CDIM) return;                      // no barriers in this kernel
  float* e  = se[t];
  float* h  = sh[t];
  float* tt = st_[t];
  for (int k = 0; k < DD; ++k) e[k] = nt[(size_t)c*DD + k];
  for (int j = 0; j < DD; ++j) {
    float acc = sb1[j];
    for (int k = 0; k < DD; ++k) acc += e[k] * sW1[k*DD + j];
    h[j] = acc;
  }
  // residual layer 1
  for (int j = 0; j < DD; ++j) {
    float acc = r1b1[j];
    for (int k = 0; k < DD; ++k) acc += h[k] * r1W1[k*DD + j];
    tt[j] = fmaxf(acc, 0.f);
  }
  for (int j = 0; j < DD; ++j) {
    float acc = r1b2[j];
    for (int k = 0; k < DD; ++k) acc += tt[k] * r1W2[k*DD + j];
    h[j] = fmaxf(acc, 0.f) + h[j];
  }
  // residual layer 2
  for (int j = 0; j < DD; ++j) {
    float acc = r2b1[j];
    for (int k = 0; k < DD; ++k) acc += h[k] * r2W1[k*DD + j];
    tt[j] = fmaxf(acc, 0.f);
  }
  for (int j = 0; j < DD; ++j) {
    float acc = r2b2[j];
    for (int k = 0; k < DD; ++k) acc += tt[k] * r2W2[k*DD + j];
    h[j] = fmaxf(acc, 0.f) + h[j];
  }
  // split head: log-softmax over 2
  float z0 = sb2[0], z1 = sb2[1];
  for (int k = 0; k < DD; ++k) { z0 += h[k] * sW2[k*2 + 0]; z1 += h[k] * sW2[k*2 + 1]; }
  const float mm = fmaxf(z0, z1);
  const float ls = mm + __logf(__expf(z0 - mm) + __expf(z1 - mm));
  ws[SPLIT1_OFF + c] = z1 - ls;
  if (c < NF) {
    const float s0 = z0 - ls;
    const float* rl = ws + RULE_OFF;
    for (int i = 0; i < NF; ++i) {
      ws[GA_OFF + c*NF + i] = rl[c*72 + i]      + s0;   // G_larg[res][i]
      ws[GB_OFF + c*NF + i] = rl[c*72 + NF + i] + s0;   // G_rarg[res][i]
    }
  }
}

// ---------------------------------------------------------------------------
// K2: vocab GEMM + online logsumexp over V (flash-style), WMMA f32 16x16x4.
// grid = (163 row-tiles, NCHUNK col chunks), 4 waves/block.
// M=16 per block: fits the register budget -> compiler software-pipelines the
// B-tile loads against the WMMA chain without spilling (verified in round 1).
// A-layout (16x4 f32): lane<16 -> M=lane, {K=4t,4t+1}; lane>=16 -> M=lane-16, {K=4t+2,4t+3}
// B-layout (4x16 f32): mirrored K pattern, N = lane%16
// D-layout: M = r + 8*(lane/16), N = lane%16
// ---------------------------------------------------------------------------
__global__ void k_vocab_lse(const float* __restrict__ nt, const float* __restrict__ vw,
                            const float* __restrict__ vb,
                            float* __restrict__ pm, float* __restrict__ ps) {
  const int mt = blockIdx.x, chunk = blockIdx.y;
  const int tid = threadIdx.x, wave = tid >> 5, lane = tid & 31;
  const int lh = lane & 15, hi = lane >> 4;
  const int row = mt * 16 + lh;
  v2f a[16];
#pragma unroll
  for (int t = 0; t < 16; ++t) {
    const int kb = 4*t + 2*hi;
    float x0 = 0.f, x1 = 0.f;
    if (row < CDIM) { x0 = nt[(size_t)row*DD + kb]; x1 = nt[(size_t)row*DD + kb + 1]; }
    a[t].x = x0; a[t].y = x1;
  }
  float m[8], s[8];
#pragma unroll
  for (int r = 0; r < 8; ++r) { m[r] = -3.0e38f; s[r] = 0.f; }

  const int ctEnd = chunk * CT_PER_CHUNK + CT_PER_CHUNK;
  for (int ct = chunk * CT_PER_CHUNK + wave; ct < ctEnd; ct += 4) {
    const int col = ct * 16 + lh;
    if (ct + 4 < ctEnd)
      __builtin_prefetch(&vw[(size_t)(2*hi)*VV + (ct + 4)*16 + lh], 0, 1);
    v8f acc = {};
#pragma unroll
    for (int t = 0; t < 16; ++t) {
      const int kb = 4*t + 2*hi;
      v2f bm;
      bm.x = vw[(size_t)kb*VV + col];
      bm.y = vw[(size_t)(kb + 1)*VV + col];
      acc = __builtin_amdgcn_wmma_f32_16x16x4_f32(false, a[t], false, bm, (short)0, acc, false, false);
    }
    const float bias = vb[col];
#pragma unroll
    for (int r = 0; r < 8; ++r) {
      const float v  = acc[r] + bias;
      const float nm = fmaxf(m[r], v);
      s[r] = s[r] * __expf(m[r] - nm) + __expf(v - nm);
      m[r] = nm;
    }
  }
  // butterfly within each 16-lane half (same rows per half under wave32)
#pragma unroll
  for (int r = 0; r < 8; ++r) {
#pragma unroll
    for (int msk = 1; msk < 16; msk <<= 1) {
      const float om = __shfl_xor(m[r], msk, 32);
      const float os = __shfl_xor(s[r], msk, 32);
      const float nm = fmaxf(m[r], om);
      s[r] = s[r] * __expf(m[r] - nm) + os * __expf(om - nm);
      m[r] = nm;
    }
  }
  __shared__ float lm[4][16], lsum[4][16];
  if (lh == 0) {
#pragma unroll
    for (int r = 0; r < 8; ++r) { lm[wave][hi*8 + r] = m[r]; lsum[wave][hi*8 + r] = s[r]; }
  }
  __syncthreads();
  if (tid < 16) {
    float mm = lm[0][tid], ss = lsum[0][tid];
    for (int w = 1; w < 4; ++w) {
      const float om = lm[w][tid], os = lsum[w][tid];
      const float nm = fmaxf(mm, om);
      ss = ss * __expf(mm - nm) + os * __expf(om - nm);
      mm = nm;
    }
    pm[(size_t)(mt*16 + tid)*NCHUNK + chunk] = mm;
    ps[(size_t)(mt*16 + tid)*NCHUNK + chunk] = ss;
  }
}

// K3: combine column-chunk partials into lse[c]
__global__ void k_lsec(const float* __restrict__ pm, const float* __restrict__ ps,
                       float* __restrict__ lse) {
  const int c = blockIdx.x * 256 + threadIdx.x;
  if (c >= CDIM) return;
  float m = pm[(size_t)c*NCHUNK], s = ps[(size_t)c*NCHUNK];
  for (int w = 1; w < NCHUNK; ++w) {
    const float om = pm[(size_t)c*NCHUNK + w], os = ps[(size_t)c*NCHUNK + w];
    const float nm = fmaxf(m, om);
    s = s * __expf(m - nm) + os * __expf(om - nm);
    m = nm;
  }
  lse[c] = m + __logf(s);
}

// ---------------------------------------------------------------------------
// K4: gathered emission GEMM: beta1[j][c] = <nt[c], vw[:,x[j]]> + vb - lse + split1
// grid = (163 row-tiles, 64 token-tiles), one wave/block, WMMA f32 16x16x4.
// ---------------------------------------------------------------------------
__global__ void k_emis(const float* __restrict__ nt, const float* __restrict__ vw,
                       const float* __restrict__ vb, const int* __restrict__ x,
                       const float* __restrict__ lse, const float* __restrict__ split1,
                       float* __restrict__ beta1) {
  const int mt = blockIdx.x, ctile = blockIdx.y;
  const int lane = threadIdx.x & 31;
  const int lh = lane & 15, hi = lane >> 4;
  const int row = mt * 16 + lh;
  v2f a[16];
#pragma unroll
  for (int t = 0; t < 16; ++t) {
    const int kb = 4*t + 2*hi;
    float x0 = 0.f, x1 = 0.f;
    if (row < CDIM) { x0 = nt[(size_t)row*DD + kb]; x1 = nt[(size_t)row*DD + kb + 1]; }
    a[t].x = x0; a[t].y = x1;
  }
  const int j   = ctile * 16 + lh;
  const int tok = x[j];
  v8f acc = {};
#pragma unroll
  for (int t = 0; t < 16; ++t) {
    const int kb = 4*t + 2*hi;
    v2f bm;
    bm.x = vw[(size_t)kb*VV + tok];
    bm.y = vw[(size_t)(kb + 1)*VV + tok];
    acc = __builtin_amdgcn_wmma_f32_16x16x4_f32(false, a[t], false, bm, (short)0, acc, false, false);
  }
  const float bias = vb[tok];
#pragma unroll
  for (int r = 0; r < 8; ++r) {
    const int rr = mt*16 + r + 8*hi;
    if (rr < CDIM)
      beta1[(size_t)j*CDIM + rr] = acc[r] + bias - lse[rr] + split1[rr];
  }
}

// K5: chart[1][n][b][res] = beta1[(b*32+n)][res]
__global__ void k_chart1(const float* __restrict__ beta1, float* __restrict__ chart) {
  const int n = blockIdx.x, b = blockIdx.y, t = threadIdx.x;
  if (t < NF) chart[CIDX(1, n, b, t)] = beta1[(size_t)(b*32 + n)*CDIM + t];
}

// ---------------------------------------------------------------------------
// K6: one CKY step for span length L. grid=(S=33-L, 32 batch), 64 threads.
// Online logsumexp over (k, i, {A,B}); tables staged in LDS.
// ---------------------------------------------------------------------------
__global__ void k_cky(const int L, const float* __restrict__ beta1,
                      const int* __restrict__ lf, const int* __restrict__ rf,
                      float* __restrict__ chart,
                      const float* __restrict__ GA, const float* __restrict__ GB) {
  const int s = blockIdx.x, b = blockIdx.y, t = threadIdx.x;
  __shared__ float sGA[NF*NF], sGB[NF*NF];
  __shared__ int   sLF[NF*NF], sRF[NF*NF];
  __shared__ float Lr[NF], Rr[NF];
  for (int i = t; i < NF*NF; i += 64) {
    sGA[i] = GA[i]; sGB[i] = GB[i];
    sLF[i] = lf[i]; sRF[i] = rf[i];
  }
  const int res = t;
  float m = -3.0e38f, ssum = 0.f;
  const float* bL = beta1 + (size_t)(b*32 + s) * CDIM;
  for (int k = 1; k < L; ++k) {
    __syncthreads();
    if (t < NF) {
      Lr[t] = chart[CIDX(k,     s,     b, t)];
      Rr[t] = chart[CIDX(L - k, s + k, b, t)];
    }
    __syncthreads();
    if (t < NF) {
      const bool rlex = (L - k) == 1;
      const bool llex = (k == 1);
      const float* bR = beta1 + (size_t)(b*32 + s + k) * CDIM;
      for (int i = 0; i < NF; ++i) {
        const int rid = sRF[i*NF + res];
        const float vA = Lr[i] + sGA[res*NF + i] +
                         (rlex ? bR[rid] : (rid < NF ? Rr[rid] : NEGV));
        const int lid = sLF[i*NF + res];
        const float vB = Rr[i] + sGB[res*NF + i] +
                         (llex ? bL[lid] : (lid < NF ? Lr[lid] : NEGV));
        float nm = fmaxf(m, vA);
        ssum = ssum * __expf(m - nm) + __expf(vA - nm); m = nm;
        nm = fmaxf(m, vB);
        ssum = ssum * __expf(m - nm) + __expf(vB - nm); m = nm;
      }
    }
  }
  if (t < NF) chart[CIDX(L, s, b, res)] = m + __logf(ssum);
}

// K7: out[b] = -logsumexp_res(chart[32][0][b][res] + root_scores[res])
__global__ void k_root(const float* __restrict__ chart, const float* __restrict__ root36,
                       float* __restrict__ out) {
  const int b = threadIdx.x;
  if (b >= 32) return;
  float m = -3.0e38f;
  for (int r = 0; r < NF; ++r) m = fmaxf(m, chart[CIDX(32, 0, b, r)] + root36[r]);
  float s = 0.f;
  for (int r = 0; r < NF; ++r) s += __expf(chart[CIDX(32, 0, b, r)] + root36[r] - m);
  out[b] = -(m + __logf(s));
}

// ---------------------------------------------------------------------------
extern "C" void kernel_launch(void* const* d_in, const int* in_sizes, int n_in,
                              void* d_out, int out_size, void* d_ws, size_t ws_size,
                              hipStream_t stream) {
  const int*   x     = (const int*)d_in[0];
  const int*   lf    = (const int*)d_in[1];
  const int*   rf    = (const int*)d_in[2];
  const float* rootW = (const float*)d_in[3];
  const float* rootb = (const float*)d_in[4];
  const float* ruleW = (const float*)d_in[5];
  const float* ruleb = (const float*)d_in[6];
  const float* nt    = (const float*)d_in[7];
  const float* sW1   = (const float*)d_in[8];
  const float* sb1   = (const float*)d_in[9];
  const float* r1W1  = (const float*)d_in[10];
  const float* r1b1  = (const float*)d_in[11];
  const float* r1W2  = (const float*)d_in[12];
  const float* r1b2  = (const float*)d_in[13];
  const float* r2W1  = (const float*)d_in[14];
  const float* r2b1  = (const float*)d_in[15];
  const float* r2W2  = (const float*)d_in[16];
  const float* r2b2  = (const float*)d_in[17];
  const float* sW2   = (const float*)d_in[18];
  const float* sb2   = (const float*)d_in[19];
  const float* vw    = (const float*)d_in[20];
  const float* vb    = (const float*)d_in[21];
  float* ws  = (float*)d_ws;
  float* out = (float*)d_out;

  k_small<<<1, 128, 0, stream>>>(rootW, rootb, ruleW, ruleb, ws);
  k_mlp<<<(CDIM + 63)/64, 64, 0, stream>>>(nt, sW1, sb1, r1W1, r1b1, r1W2, r1b2,
                                           r2W1, r2b1, r2W2, r2b2, sW2, sb2, ws);
  k_vocab_lse<<<dim3(MTILES, NCHUNK), 128, 0, stream>>>(nt, vw, vb, ws + PM_OFF, ws + PS_OFF);
  k_lsec<<<(CDIM + 255)/256, 256, 0, stream>>>(ws + PM_OFF, ws + PS_OFF, ws + LSE_OFF);
  k_emis<<<dim3(MTILES, 64), 32, 0, stream>>>(nt, vw, vb, x, ws + LSE_OFF,
                                              ws + SPLIT1_OFF, ws + BETA_OFF);
  k_chart1<<<dim3(32, 32), 64, 0, stream>>>(ws + BETA_OFF, ws + CHART_OFF);
  for (int L = 2; L <= 32; ++L)
    k_cky<<<dim3(33 - L, 32), 64, 0, stream>>>(L, ws + BETA_OFF, lf, rf,
                                               ws + CHART_OFF, ws + GA_OFF, ws + GB_OFF);
  k_root<<<1, 32, 0, stream>>>(ws + CHART_OFF, ws + ROOT_OFF, out);
}